// ADSAGE_7232724927264
// MI455X (gfx1250) — compile-verified
//
#include <hip/hip_runtime.h>
#include <hip/hip_bf16.h>

typedef __attribute__((ext_vector_type(2))) float v2f;
typedef __attribute__((ext_vector_type(8))) float v8f;

#define TPB 256

// ---------------------------------------------------------------------------
// Degree count: cnt[dst] += 1 for each edge (float, exact up to 2^24)
// ---------------------------------------------------------------------------
__global__ void gnn_count_kernel(const long long* __restrict__ dst,
                                 float* __restrict__ cnt, int E) {
    int e = blockIdx.x * blockDim.x + threadIdx.x;
    if (e < E) unsafeAtomicAdd(&cnt[(int)dst[e]], 1.0f);
}

// invcnt = 1/max(cnt,1) (SAGE mean), dinv = rsqrt(cnt+1) (GCN with self-loop)
__global__ void gnn_degprep_kernel(const float* __restrict__ cnt,
                                   float* __restrict__ invcnt,
                                   float* __restrict__ dinv, int n) {
    int i = blockIdx.x * blockDim.x + threadIdx.x;
    if (i >= n) return;
    float c = cnt[i];
    invcnt[i] = 1.0f / fmaxf(c, 1.0f);
    dinv[i]   = rsqrtf(c + 1.0f);
}

// ---------------------------------------------------------------------------
// Edge scatter-add: out[dst[e]][:] += feat[src[e]][:]
// grid = (ceil(E/TPB), F/4); each thread moves 4 floats of one edge
// ---------------------------------------------------------------------------
__global__ void gnn_scatter_kernel(const float* __restrict__ feat,
                                   const long long* __restrict__ src,
                                   const long long* __restrict__ dst,
                                   float* __restrict__ out, int E, int F) {
    int e = blockIdx.x * blockDim.x + threadIdx.x;
    if (e >= E) return;
    int g4 = blockIdx.y << 2;
    const float* f = feat + (size_t)(int)src[e] * F + g4;
    float*       o = out  + (size_t)(int)dst[e] * F + g4;
    float4 v = *(const float4*)f;
    unsafeAtomicAdd(o + 0, v.x);
    unsafeAtomicAdd(o + 1, v.y);
    unsafeAtomicAdd(o + 2, v.z);
    unsafeAtomicAdd(o + 3, v.w);
}

// ---------------------------------------------------------------------------
// Fused WMMA GEMM (fp32, V_WMMA_F32_16X16X4_F32):
//   out = act( (A1 .* ascale_row) @ W1  [+ A2 @ W2]  [+ bias] ) [.* oscale_row]
// One wave per 16x16 output tile. grid = (Fout/16, N/16), block = 32.
// VGPR layouts per CDNA5 ISA 7.12.2:
//   A 16x4 f32: lanes 0-15 -> {K=0,K=1}, lanes 16-31 -> {K=2,K=3}
//   B  4x16 f32 (mirrored): lanes 0-15 rows {0,1}, lanes 16-31 rows {2,3}
//   C/D 16x16 f32: VGPR r, lanes 0-15: M=r; lanes 16-31: M=8+r
// ---------------------------------------------------------------------------
__global__ void gnn_wmma_gemm_kernel(const float* __restrict__ A1,
                                     const float* __restrict__ ascale,
                                     const float* __restrict__ W1,
                                     const float* __restrict__ A2,
                                     const float* __restrict__ W2,
                                     const float* __restrict__ bias,
                                     const float* __restrict__ oscale,
                                     float* __restrict__ out,
                                     int Fin, int Fout, int relu) {
    const int lane = threadIdx.x & 31;
    const int half = lane >> 4;      // selects K pair (A/B) and M block (C/D)
    const int l16  = lane & 15;
    const int row0 = blockIdx.y << 4;
    const int col0 = blockIdx.x << 4;

    v8f c = {};

    const float sc1 = ascale ? ascale[row0 + l16] : 1.0f;
    const float* a1row = A1 + (size_t)(row0 + l16) * Fin + half * 2;
    const float* w1p   = W1 + col0 + l16;
    for (int k = 0; k < Fin; k += 4) {
        v2f a, b;
        float2 av = *(const float2*)(a1row + k);
        a.x = av.x * sc1;
        a.y = av.y * sc1;
        const float* w = w1p + (size_t)(k + half * 2) * Fout;
        b.x = w[0];
        b.y = w[Fout];
        c = __builtin_amdgcn_wmma_f32_16x16x4_f32(false, a, false, b,
                                                  (short)0, c, false, false);
    }

    if (A2) {
        const float* a2row = A2 + (size_t)(row0 + l16) * Fin + half * 2;
        const float* w2p   = W2 + col0 + l16;
        for (int k = 0; k < Fin; k += 4) {
            v2f a, b;
            float2 av = *(const float2*)(a2row + k);
            a.x = av.x;
            a.y = av.y;
            const float* w = w2p + (size_t)(k + half * 2) * Fout;
            b.x = w[0];
            b.y = w[Fout];
            c = __builtin_amdgcn_wmma_f32_16x16x4_f32(false, a, false, b,
                                                      (short)0, c, false, false);
        }
    }

    const float bc = bias ? bias[col0 + l16] : 0.0f;
#pragma unroll
    for (int r = 0; r < 8; ++r) {
        int row = row0 + half * 8 + r;
        float v = c[r] + bc;
        if (relu) v = fmaxf(v, 0.0f);
        if (oscale) v *= oscale[row];
        out[(size_t)row * Fout + col0 + l16] = v;
    }
}

// ---------------------------------------------------------------------------
// Row softmax over 64 columns; one wave (32 lanes) per node, 2 cols per lane
// ---------------------------------------------------------------------------
__global__ void gnn_softmax64_kernel(float* __restrict__ h, int n) {
    int wave = threadIdx.x >> 5;
    int lane = threadIdx.x & 31;
    int node = blockIdx.x * (blockDim.x >> 5) + wave;
    if (node >= n) return;
    float* row = h + (size_t)node * 64;
    float a = row[lane];
    float b = row[lane + 32];
    float m = fmaxf(a, b);
#pragma unroll
    for (int off = 16; off > 0; off >>= 1) m = fmaxf(m, __shfl_xor(m, off, 32));
    float ea = __expf(a - m);
    float eb = __expf(b - m);
    float s = ea + eb;
#pragma unroll
    for (int off = 16; off > 0; off >>= 1) s += __shfl_xor(s, off, 32);
    float inv = 1.0f / s;
    row[lane]      = ea * inv;
    row[lane + 32] = eb * inv;
}

// out[i][c] = (acc[i][c] + y[i][c]) * dinv[i] + bg[c]     (C = 64)
__global__ void gnn_gcn_final_kernel(const float* __restrict__ acc,
                                     const float* __restrict__ y,
                                     const float* __restrict__ dinv,
                                     const float* __restrict__ bg,
                                     float* __restrict__ out, int total) {
    int idx = blockIdx.x * blockDim.x + threadIdx.x;
    if (idx >= total) return;
    int i = idx >> 6;
    int col = idx & 63;
    out[idx] = (acc[idx] + y[idx]) * dinv[i] + bg[col];
}

// ---------------------------------------------------------------------------
extern "C" void kernel_launch(void* const* d_in, const int* in_sizes, int n_in,
                              void* d_out, int out_size, void* d_ws, size_t ws_size,
                              hipStream_t stream) {
    const int FIN = 128, FH = 128, FC = 64;
    const int N = in_sizes[0] / FIN;   // 50000 (multiple of 16)
    const int E = in_sizes[1] / 2;     // 800000

    const float*     x    = (const float*)d_in[0];
    const long long* ei   = (const long long*)d_in[1];
    const long long* srcI = ei;
    const long long* dstI = ei + E;
    const float* W1l = (const float*)d_in[2];
    const float* b1l = (const float*)d_in[3];
    const float* W1r = (const float*)d_in[4];
    const float* W2l = (const float*)d_in[5];
    const float* b2l = (const float*)d_in[6];
    const float* W2r = (const float*)d_in[7];
    const float* Wg  = (const float*)d_in[8];
    const float* bg  = (const float*)d_in[9];
    float* out = (float*)d_out;

    // ---- workspace layout (≈65 MB) ----
    char* w = (char*)d_ws;
    size_t npad = (((size_t)N * 4) + 255) & ~(size_t)255;
    float* cnt    = (float*)w; w += npad;
    float* invcnt = (float*)w; w += npad;
    float* dinv   = (float*)w; w += npad;
    float* agg    = (float*)w; w += (size_t)N * FIN * 4;  // agg1 / agg2 / gcn acc
    float* h1     = (float*)w; w += (size_t)N * FH  * 4;  // h1, later reused as y
    float* h2     = (float*)w; w += (size_t)N * FC  * 4;  // h2 -> softmax in place

    const int eBlocks = (E + TPB - 1) / TPB;
    const int nBlocks = (N + TPB - 1) / TPB;
    const int rowT = N / 16;

    // degrees
    hipMemsetAsync(cnt, 0, (size_t)N * 4, stream);
    gnn_count_kernel<<<eBlocks, TPB, 0, stream>>>(dstI, cnt, E);
    gnn_degprep_kernel<<<nBlocks, TPB, 0, stream>>>(cnt, invcnt, dinv, N);

    // ---- SAGE layer 1: h1 = relu(mean(x)@W1l + x@W1r + b1l) ----
    hipMemsetAsync(agg, 0, (size_t)N * FIN * 4, stream);
    gnn_scatter_kernel<<<dim3(eBlocks, FIN / 4), TPB, 0, stream>>>(x, srcI, dstI, agg, E, FIN);
    gnn_wmma_gemm_kernel<<<dim3(FH / 16, rowT), 32, 0, stream>>>(
        agg, invcnt, W1l, x, W1r, b1l, nullptr, h1, FIN, FH, 1);

    // ---- SAGE layer 2: h2 = mean(h1)@W2l + h1@W2r + b2l ----
    hipMemsetAsync(agg, 0, (size_t)N * FH * 4, stream);
    gnn_scatter_kernel<<<dim3(eBlocks, FH / 4), TPB, 0, stream>>>(h1, srcI, dstI, agg, E, FH);
    gnn_wmma_gemm_kernel<<<dim3(FC / 16, rowT), 32, 0, stream>>>(
        agg, invcnt, W2l, h1, W2r, b2l, nullptr, h2, FH, FC, 0);

    // ---- softmax over 64 cols (in place) ----
    gnn_softmax64_kernel<<<(N + 7) / 8, TPB, 0, stream>>>(h2, N);

    // ---- GCN: y = (h3 @ Wg) * dinv[row]   (y stored in h1 buffer) ----
    gnn_wmma_gemm_kernel<<<dim3(FC / 16, rowT), 32, 0, stream>>>(
        h2, nullptr, Wg, nullptr, nullptr, nullptr, dinv, h1, FC, FC, 0);

    // acc[dst] += y[src]  (acc in agg buffer, N x 64)
    hipMemsetAsync(agg, 0, (size_t)N * FC * 4, stream);
    gnn_scatter_kernel<<<dim3(eBlocks, FC / 4), TPB, 0, stream>>>(h1, srcI, dstI, agg, E, FC);

    // out = (acc + y) * dinv + bg
    int total = N * FC;
    gnn_gcn_final_kernel<<<(total + TPB - 1) / TPB, TPB, 0, stream>>>(
        agg, h1, dinv, bg, out, total);
}